// NanEmbedOld_25228637896776
// MI455X (gfx1250) — compile-verified
//
#include <hip/hip_runtime.h>
#include <hip/hip_bf16.h>

typedef float v2f  __attribute__((ext_vector_type(2)));
typedef float v8f  __attribute__((ext_vector_type(8)));
typedef unsigned int v4u __attribute__((ext_vector_type(4)));
typedef int   v8i  __attribute__((ext_vector_type(8)));
typedef int   v4i  __attribute__((ext_vector_type(4)));

#define NROWS 8192
#define FIN   256
#define DOUT  64

// out[n,d] = ( (x @ W)[n,d] + sum_f b[f,d] ) * (1/256)
// Grid: 64 blocks x 256 threads (8 waves). Wave w of block bk owns rows
// (bk*8+w)*16 .. +15 and computes the full 16x64 tile with f32 WMMA.
__global__ __launch_bounds__(256)
void perfeat_linear_mean_wmma(const float* __restrict__ x,
                              const float* __restrict__ W,
                              const float* __restrict__ b,
                              float* __restrict__ out)
{
    __shared__ float Wlds[FIN * DOUT];   // 64 KB: whole W, TDM-resident
    __shared__ float bpart[4 * DOUT];    // bias partial sums

    const int tid  = threadIdx.x;
    const int wave = tid >> 5;
    const int lane = tid & 31;
    const int half = lane >> 4;   // 0: K pair {0,1}; 1: K pair {2,3}
    const int l15  = lane & 15;

    // ---- Tensor Data Mover: DMA W[256][64] f32 -> LDS (wave 0 only) ----
    if (wave == 0) {
        unsigned lds_off = (unsigned)(size_t)(&Wlds[0]);
        unsigned long long ga = (unsigned long long)(size_t)W;
        v4u g0;
        g0[0] = 1u;                                   // count=1 (valid user D#)
        g0[1] = lds_off;                              // lds_addr
        g0[2] = (unsigned)(ga & 0xFFFFFFFFu);         // global_addr[31:0]
        g0[3] = (unsigned)((ga >> 32) & 0x1FFFFFFu)   // global_addr[56:32]
              | (2u << 30);                           // type=2 (image)
        v8i g1;
        g1[0] = (int)(2u << 16);        // workgroup_mask=0, data_size=2 (4B)
        g1[1] = (int)(64u  << 16);      // tensor_dim0[15:0]=64 (cols)
        g1[2] = (int)(256u << 16);      // dim0 hi=0 | tensor_dim1[15:0]=256 (rows)
        g1[3] = (int)(64u  << 16);      // dim1 hi=0 | tile_dim0=64
        g1[4] = 256;                    // tile_dim1=256, tile_dim2=0
        g1[5] = 64;                     // tensor_dim0_stride=64 (f32 units)
        g1[6] = 0;                      // stride0 hi | tensor_dim1_stride lo (unused, 2D)
        g1[7] = 0;
        v4i z4 = {0, 0, 0, 0};
        v8i z8 = {0, 0, 0, 0, 0, 0, 0, 0};
        // amdgpu-toolchain (clang-23 / therock-10.0) 6-arg form
        __builtin_amdgcn_tensor_load_to_lds(g0, g1, z4, z4, z8, 0);
    }

    // ---- bias partial sums, overlapped with the TDM transfer ----
    {
        const int d = tid & 63;
        const int g = tid >> 6;                 // 4 groups of 64 f-rows
        const float* bp = b + (g * 64) * DOUT + d;
        float s = 0.0f;
        #pragma unroll 8
        for (int i = 0; i < 64; ++i) s += bp[i * DOUT];
        bpart[g * 64 + d] = s;
    }

    if (wave == 0) __builtin_amdgcn_s_wait_tensorcnt(0);
    __syncthreads();

    // ---- accumulators seeded with the bias column sums (C-matrix init) ----
    v8f acc[4];
    #pragma unroll
    for (int j = 0; j < 4; ++j) {
        const int col = j * 16 + l15;           // N index for this lane
        float s = bpart[col] + bpart[64 + col] + bpart[128 + col] + bpart[192 + col];
        #pragma unroll
        for (int r = 0; r < 8; ++r) acc[j][r] = s;
    }

    // ---- main K loop: 64 steps of K=4, f32 WMMA 16x16x4 ----
    const int m0  = (blockIdx.x * 8 + wave) * 16;
    const float* xrow = x + (m0 + l15) * FIN + half * 2;  // A: lane = M row

    #pragma unroll 4
    for (int kk = 0; kk < 64; ++kk) {
        v2f a = *(const v2f*)(xrow + kk * 4);   // K = 4kk + 2*half + {0,1}
        const int krow = kk * 4 + half * 2;
        const float* wl = &Wlds[krow * DOUT + l15];
        #pragma unroll
        for (int j = 0; j < 4; ++j) {
            v2f bb;
            bb[0] = wl[j * 16];                 // B: K=krow,   N=16j+l15
            bb[1] = wl[DOUT + j * 16];          // B: K=krow+1, N=16j+l15
            acc[j] = __builtin_amdgcn_wmma_f32_16x16x4_f32(
                false, a, false, bb, (short)0, acc[j], false, false);
        }
    }

    // ---- epilogue: mean scale + store (C/D layout: VGPR r -> M = r + 8*half) ----
    float* orow = out + (m0 + 8 * half) * DOUT + l15;
    #pragma unroll
    for (int j = 0; j < 4; ++j) {
        #pragma unroll
        for (int r = 0; r < 8; ++r) {
            orow[r * DOUT + j * 16] = acc[j][r] * (1.0f / 256.0f);
        }
    }
}

extern "C" void kernel_launch(void* const* d_in, const int* in_sizes, int n_in,
                              void* d_out, int out_size, void* d_ws, size_t ws_size,
                              hipStream_t stream) {
    const float* x = (const float*)d_in[0];   // [8192, 256]
    const float* W = (const float*)d_in[1];   // [256, 64]
    const float* b = (const float*)d_in[2];   // [256, 64]
    float* out = (float*)d_out;               // [8192, 64]
    dim3 grid(NROWS / 128);                   // 8 waves * 16 rows per block
    dim3 block(256);
    perfeat_linear_mean_wmma<<<grid, block, 0, stream>>>(x, W, b, out);
}